// Encoder_Conv_25494925869659
// MI455X (gfx1250) — compile-verified
//
#include <hip/hip_runtime.h>
#include <hip/hip_bf16.h>
#include <math.h>

typedef __attribute__((ext_vector_type(2))) float v2f;
typedef __attribute__((ext_vector_type(4))) float v4f;
typedef __attribute__((ext_vector_type(8))) float v8f;

#define IN_EPS 1e-5f
#define LSLOPE 0.4f

__device__ __forceinline__ float leaky(float v) { return v >= 0.f ? v : LSLOPE * v; }

__device__ __forceinline__ v8f wmma_f32(v2f a, v2f b, v8f c) {
  // D = A(16x4,f32) * B(4x16,f32) + C ; emits v_wmma_f32_16x16x4_f32
  return __builtin_amdgcn_wmma_f32_16x16x4_f32(false, a, false, b, (short)0, c,
                                               false, false);
}

// ---------------------------------------------------------------------------
// Stage 1: h1(4,4096) = leaky( x(4,40960) @ W_fc^T + b_fc )
// Memory-bound (2 FLOP/byte): one block per output column n streams one W row
// (160KB) once with NON-TEMPORAL b128 loads (671MB single-use stream must not
// rinse L2); x (640KB) stays L2-resident and is re-read per block.
// Roofline: 671MB / 23.3TB/s ~= 29us -> dominates the whole graph.
// ---------------------------------------------------------------------------
__global__ void __launch_bounds__(256) fc_kernel(const float* __restrict__ x,
                                                 const float* __restrict__ W,
                                                 const float* __restrict__ b,
                                                 float* __restrict__ out) {
  const int n = blockIdx.x;
  const int tid = threadIdx.x;
  const v4f* Wr = (const v4f*)(W + (size_t)n * 40960);
  const v4f* x0 = (const v4f*)(x);
  const v4f* x1 = (const v4f*)(x + 40960);
  const v4f* x2 = (const v4f*)(x + 81920);
  const v4f* x3 = (const v4f*)(x + 122880);
  float a0 = 0.f, a1 = 0.f, a2 = 0.f, a3 = 0.f;
  for (int i = tid; i < 10240; i += 256) {
    v4f w4 = __builtin_nontemporal_load(&Wr[i]);   // TH=NT stream
    v4f v;
    v = x0[i]; a0 += w4.x * v.x + w4.y * v.y + w4.z * v.z + w4.w * v.w;
    v = x1[i]; a1 += w4.x * v.x + w4.y * v.y + w4.z * v.z + w4.w * v.w;
    v = x2[i]; a2 += w4.x * v.x + w4.y * v.y + w4.z * v.z + w4.w * v.w;
    v = x3[i]; a3 += w4.x * v.x + w4.y * v.y + w4.z * v.z + w4.w * v.w;
  }
  __shared__ v4f red[256];
  v4f mine; mine.x = a0; mine.y = a1; mine.z = a2; mine.w = a3;
  red[tid] = mine;
  __syncthreads();
  for (int s = 128; s > 0; s >>= 1) {
    if (tid < s) red[tid] = red[tid] + red[tid + s];
    __syncthreads();
  }
  if (tid == 0) {
    v4f r = red[0];
    float bb = b[n];
    out[n]         = leaky(r.x + bb);
    out[4096 + n]  = leaky(r.y + bb);
    out[8192 + n]  = leaky(r.z + bb);
    out[12288 + n] = leaky(r.w + bb);
  }
}

// ---------------------------------------------------------------------------
// Stage 2: fused 3x3 conv (pad=1) + InstanceNorm (affine=False).
// One block per (n, co) output plane; plane held in LDS; block reduce mean/var.
// ---------------------------------------------------------------------------
__global__ void __launch_bounds__(256) conv_in_kernel(
    const float* __restrict__ in, float* __restrict__ out,
    const float* __restrict__ w, const float* __restrict__ bias,
    int N, int Cin, int Cout, int Hin, int Win, int stride) {
  const int n  = blockIdx.x / Cout;
  const int co = blockIdx.x % Cout;
  const int Hout = (Hin - 1) / stride + 1;   // = floor((Hin+2-3)/s)+1
  const int Wout = (Win - 1) / stride + 1;
  const int cnt = Hout * Wout;               // <= 4096
  __shared__ float plane[4096];
  __shared__ float red1[256];
  __shared__ float red2[256];
  const int tid = threadIdx.x;
  float s1 = 0.f, s2 = 0.f;
  for (int p = tid; p < cnt; p += 256) {
    int oh = p / Wout, ow = p % Wout;
    float acc = bias[co];
    for (int ci = 0; ci < Cin; ++ci) {
      const float* ip = in + (size_t)(n * Cin + ci) * Hin * Win;
      const float* wp = w + (size_t)(co * Cin + ci) * 9;
      for (int kh = 0; kh < 3; ++kh) {
        int ih = oh * stride - 1 + kh;
        if (ih < 0 || ih >= Hin) continue;
        for (int kw = 0; kw < 3; ++kw) {
          int iw = ow * stride - 1 + kw;
          if (iw < 0 || iw >= Win) continue;
          acc += ip[ih * Win + iw] * wp[kh * 3 + kw];
        }
      }
    }
    plane[p] = acc;
    s1 += acc;
    s2 += acc * acc;
  }
  red1[tid] = s1;
  red2[tid] = s2;
  __syncthreads();
  for (int s = 128; s > 0; s >>= 1) {
    if (tid < s) { red1[tid] += red1[tid + s]; red2[tid] += red2[tid + s]; }
    __syncthreads();
  }
  float mean = red1[0] / (float)cnt;
  float var  = red2[0] / (float)cnt - mean * mean;
  float inv  = rsqrtf(var + IN_EPS);
  float* op = out + (size_t)(n * Cout + co) * cnt;
  for (int p = tid; p < cnt; p += 256) op[p] = (plane[p] - mean) * inv;
}

// ---------------------------------------------------------------------------
// Stage 3: 64-layer tanh RNN (T=2, B=4, H=64) + output head, one workgroup.
// 4 waves; wave w owns output columns [16w, 16w+16). GEMMs via
// v_wmma_f32_16x16x4_f32 with M padded 8(or 4)->16 with zero rows.
// ISA f32 fragment layouts (CDNA5 ISA 7.12.2):
//   A 16x4 : lane<16 -> {A[r][k],A[r][k+1]} ; lane>=16 -> {A[r][k+2],A[r][k+3]}
//   B 4x16 : lane<16 -> {B[k][n],B[k+1][n]} ; lane>=16 -> {B[k+2][n],B[k+3][n]}
//   C/D    : vgpr v, lane<16 -> M=v ; lane>=16 -> M=v+8 ; N=lane&15
// l==0 reads A from global seq (global_load path); l>=1 reads A from LDS ys
// kept zero-padded to 16 rows (pure ds_load path; no flat loads, no masking).
// ---------------------------------------------------------------------------
__global__ void __launch_bounds__(128) rnn_kernel(
    const float* __restrict__ seq,     // (8,1024) flat = (T=2,B=4,1024)
    const float* __restrict__ W_ih0,   // (64,1024)
    const float* __restrict__ b_ih0,   // (64)
    const float* __restrict__ W_ihr,   // (63,64,64)
    const float* __restrict__ b_ihr,   // (63,64)
    const float* __restrict__ W_hh,    // (64,64,64)
    const float* __restrict__ b_hh,    // (64,64)
    const float* __restrict__ W_out,   // (34,64)
    const float* __restrict__ b_out,   // (34)
    float* __restrict__ out) {         // (8,34) = (2,4,34)
  __shared__ float ys[16 * 64];   // layer activations (T*B,H); rows 8..15 zero
  __shared__ float hs[16 * 64];   // hidden state; rows 4..15 stay zero
  __shared__ float pre[8 * 64];   // W_ih @ x + b_ih

  const int tid  = threadIdx.x;
  const int lane = tid & 31;
  const int wave = tid >> 5;
  const int r    = lane & 15;       // fragment row / col-in-tile
  const int hi   = lane >> 4;       // half-wave selector
  const int kk   = hi * 2;          // K sub-offset for this half
  const int col  = wave * 16 + r;   // owned output column j in [0,64)

  for (int i = tid; i < 16 * 64; i += 128) { hs[i] = 0.f; ys[i] = 0.f; }
  __syncthreads();

  for (int l = 0; l < 64; ++l) {
    const float* bih = (l == 0) ? b_ih0 : (b_ihr + (l - 1) * 64);
    const float* Whh = W_hh + (size_t)l * 64 * 64;
    const float* bhh = b_hh + l * 64;

    // reset hidden state rows 0..3 for this layer (h0 = zeros)
    for (int i = tid; i < 4 * 64; i += 128) hs[i] = 0.f;

    // ---- pre(8x64) = In @ Wih^T + bih, per-wave 16-col tile --------------
    v8f acc = {};
    if (l == 0) {                               // uniform branch (EXEC full)
      const float* wrow = W_ih0 + (size_t)col * 1024 + kk;
      const int rr = r & 7;                     // clamp to valid rows
      const float msk = (r < 8) ? 1.f : 0.f;    // zero-pad rows 8..15
      for (int k = 0; k < 1024; k += 4) {
        const float* ap = seq + (size_t)rr * 1024 + k + kk;
        v2f a; a.x = ap[0] * msk; a.y = ap[1] * msk;
        v2f bb; bb.x = wrow[k]; bb.y = wrow[k + 1];
        acc = wmma_f32(a, bb, acc);
      }
    } else {
      const float* wrow = W_ihr + (size_t)(l - 1) * 64 * 64 + (size_t)col * 64 + kk;
      for (int k = 0; k < 64; k += 4) {
        v2f a; a.x = ys[r * 64 + k + kk]; a.y = ys[r * 64 + k + kk + 1];
        v2f bb; bb.x = wrow[k]; bb.y = wrow[k + 1];
        acc = wmma_f32(a, bb, acc);
      }
    }
    if (hi == 0) {
      float bv = bih[col];
#pragma unroll
      for (int v = 0; v < 8; ++v) pre[v * 64 + col] = acc[v] + bv;  // M=v
    }
    __syncthreads();

    // ---- recurrence: h_t = tanh(pre_t + h_{t-1} @ W_hh^T + b_hh) ---------
    for (int t = 0; t < 2; ++t) {
      v8f g = {};
      const float* wh = Whh + (size_t)col * 64 + kk;
      for (int k = 0; k < 64; k += 4) {
        v2f a; a.x = hs[r * 64 + k + kk]; a.y = hs[r * 64 + k + kk + 1];
        v2f bb; bb.x = wh[k]; bb.y = wh[k + 1];
        g = wmma_f32(a, bb, g);
      }
      float vals[4];
      if (hi == 0) {
        float bv = bhh[col];
#pragma unroll
        for (int v = 0; v < 4; ++v)   // M=v = batch b (rows 0..3 valid)
          vals[v] = tanhf(pre[(t * 4 + v) * 64 + col] + g[v] + bv);
      }
      __syncthreads();
      if (hi == 0) {
#pragma unroll
        for (int v = 0; v < 4; ++v) {
          hs[v * 64 + col] = vals[v];
          ys[(t * 4 + v) * 64 + col] = vals[v];
        }
      }
      __syncthreads();
    }
  }

  // ---- head: out(8,34) = leaky(ys @ W_out^T + b_out) ---------------------
  for (int idx = tid; idx < 8 * 34; idx += 128) {
    int m = idx / 34, j = idx % 34;
    float s = b_out[j];
    const float* wr = W_out + j * 64;
    for (int k = 0; k < 64; ++k) s += ys[m * 64 + k] * wr[k];
    out[idx] = leaky(s);
  }
}

// ---------------------------------------------------------------------------
extern "C" void kernel_launch(void* const* d_in, const int* in_sizes, int n_in,
                              void* d_out, int out_size, void* d_ws, size_t ws_size,
                              hipStream_t stream) {
  const float* x     = (const float*)d_in[0];
  const float* W_fc  = (const float*)d_in[1];
  const float* b_fc  = (const float*)d_in[2];
  const float* w0 = (const float*)d_in[3];   const float* b0 = (const float*)d_in[4];
  const float* w1 = (const float*)d_in[5];   const float* b1 = (const float*)d_in[6];
  const float* w2 = (const float*)d_in[7];   const float* b2 = (const float*)d_in[8];
  const float* w3 = (const float*)d_in[9];   const float* b3 = (const float*)d_in[10];
  const float* w4 = (const float*)d_in[11];  const float* b4 = (const float*)d_in[12];
  const float* w5 = (const float*)d_in[13];  const float* b5 = (const float*)d_in[14];
  const float* W_ih0 = (const float*)d_in[15];
  const float* b_ih0 = (const float*)d_in[16];
  const float* W_ihr = (const float*)d_in[17];
  const float* b_ihr = (const float*)d_in[18];
  const float* W_hh  = (const float*)d_in[19];
  const float* b_hh  = (const float*)d_in[20];
  const float* W_out = (const float*)d_in[21];
  const float* b_out = (const float*)d_in[22];

  float* ws   = (float*)d_ws;
  float* h1   = ws;            // 4*4096
  float* bufA = ws + 16384;    // 16384 floats
  float* bufB = ws + 32768;    // 16384 floats

  // Stage 1: dominant memory-bound GEMV (~29us roofline at 23.3 TB/s)
  fc_kernel<<<4096, 256, 0, stream>>>(x, W_fc, b_fc, h1);

  // Stage 2: conv + instance-norm chain; h1 viewed as (4,1,4,1024)
  conv_in_kernel<<<4,   256, 0, stream>>>(h1,   bufA, w0, b0, 4, 1,  1,  4, 1024, 1);
  conv_in_kernel<<<16,  256, 0, stream>>>(bufA, bufB, w1, b1, 4, 1,  4,  4, 1024, 2);
  conv_in_kernel<<<32,  256, 0, stream>>>(bufB, bufA, w2, b2, 4, 4,  8,  2, 512,  2);
  conv_in_kernel<<<64,  256, 0, stream>>>(bufA, bufB, w3, b3, 4, 8,  16, 1, 256,  2);
  conv_in_kernel<<<128, 256, 0, stream>>>(bufB, bufA, w4, b4, 4, 16, 32, 1, 128,  2);
  conv_in_kernel<<<256, 256, 0, stream>>>(bufA, bufB, w5, b5, 4, 32, 64, 1, 64,   2);

  // Stage 3: sequential 64-layer RNN + head, persistent single workgroup
  // (bufB flat (4,64,1,32) == reshape (2,4,1024))
  rnn_kernel<<<1, 128, 0, stream>>>(bufB, W_ih0, b_ih0, W_ihr, b_ihr,
                                    W_hh, b_hh, W_out, b_out, (float*)d_out);
}